// GCN_2516850835925
// MI455X (gfx1250) — compile-verified
//
#include <hip/hip_runtime.h>

typedef __attribute__((ext_vector_type(2))) float v2f;
typedef __attribute__((ext_vector_type(8))) float v8f;

#define FIN 128
#define FHID 128
#define FOUT 64

// ---------------- degree / norm ----------------
__global__ void init_deg_kernel(float* __restrict__ deg, int n) {
    int i = blockIdx.x * blockDim.x + threadIdx.x;
    if (i < n) deg[i] = 1.0f;  // self-loop contribution
}

__global__ void deg_accum_kernel(const int* __restrict__ dst, float* __restrict__ deg, int e) {
    int i = blockIdx.x * blockDim.x + threadIdx.x;
    if (i < e) atomicAdd(&deg[dst[i]], 1.0f);
}

__global__ void dinv_kernel(float* __restrict__ deg, int n) {
    int i = blockIdx.x * blockDim.x + threadIdx.x;
    if (i < n) {
        float d = deg[i];
        deg[i] = (d > 0.0f) ? rsqrtf(fmaxf(d, 1.0f)) : 0.0f;
    }
}

// ---------------- utility ----------------
__global__ void zero4_kernel(float4* __restrict__ p, int n4) {
    int i = blockIdx.x * blockDim.x + threadIdx.x;
    if (i < n4) p[i] = make_float4(0.f, 0.f, 0.f, 0.f);
}

__global__ void init_out_b2_kernel(float* __restrict__ out, const float* __restrict__ b2, int n) {
    int i = blockIdx.x * blockDim.x + threadIdx.x;
    if (i < n) out[i] = b2[i & (FOUT - 1)];
}

// ---------------- GEMM1: XW = X @ W1 (fp32 WMMA 16x16x4) ----------------
// block = 256 threads = 8 waves; each wave computes one 16-row strip x 128 cols.
__global__ void __launch_bounds__(256)
gemm_x_w1(const float* __restrict__ X, const float* __restrict__ W,
          float* __restrict__ out, int M) {
    __shared__ float ldsW[FIN * FHID];  // transposed: ldsW[n*128 + k]

    const int t = threadIdx.x;
    // cooperative load of W1 (row-major [k*128+n]) -> transposed LDS
    for (int i = t; i < (FIN * FHID) / 4; i += 256) {
        float4 w = ((const float4*)W)[i];
        int k = (i * 4) / FHID;
        int n = (i * 4) % FHID;
        ldsW[(n + 0) * FIN + k] = w.x;
        ldsW[(n + 1) * FIN + k] = w.y;
        ldsW[(n + 2) * FIN + k] = w.z;
        ldsW[(n + 3) * FIN + k] = w.w;
    }
    __syncthreads();

    const int wave = t >> 5;
    const int lane = t & 31;
    const int rowTile = blockIdx.x * 8 + wave;
    const int numRowTiles = M >> 4;  // M multiple of 16
    if (rowTile >= numRowTiles) return;

    const int m = lane & 15;   // row within tile (A) / col within tile (B,C)
    const int h = lane >> 4;   // half-wave select
    const long row = (long)rowTile * 16 + m;
    const float* xrow = X + row * FIN;

    v8f acc[8];
    v8f zf = {};
#pragma unroll
    for (int nt = 0; nt < 8; ++nt) acc[nt] = zf;

    for (int kk = 0; kk < FIN; kk += 4) {
        const int c0 = kk + 2 * h;
        v2f a;
        a.x = xrow[c0];
        a.y = xrow[c0 + 1];
#pragma unroll
        for (int nt = 0; nt < 8; ++nt) {
            v2f b;
            b.x = ldsW[(nt * 16 + m) * FIN + c0];
            b.y = ldsW[(nt * 16 + m) * FIN + c0 + 1];
            acc[nt] = __builtin_amdgcn_wmma_f32_16x16x4_f32(
                false, a, false, b, (short)0, acc[nt], false, false);
        }
    }

    // D layout: VGPR r -> rows r (lanes 0-15) and r+8 (lanes 16-31), col = m
    const long rbase = (long)rowTile * 16 + 8 * h;
#pragma unroll
    for (int nt = 0; nt < 8; ++nt) {
#pragma unroll
        for (int r = 0; r < 8; ++r) {
            out[(rbase + r) * FHID + nt * 16 + m] = acc[nt][r];
        }
    }
}

// ---------------- GEMM2: HW2 = relu(H1 + b1) @ W2 ----------------
__global__ void __launch_bounds__(256)
gemm_h_w2(const float* __restrict__ H, const float* __restrict__ b1,
          const float* __restrict__ W, float* __restrict__ out, int M) {
    __shared__ float ldsW[FOUT * FHID];  // transposed: ldsW[n*128 + k]
    __shared__ float ldsB1[FHID];

    const int t = threadIdx.x;
    for (int i = t; i < (FHID * FOUT) / 4; i += 256) {
        float4 w = ((const float4*)W)[i];  // W2 row-major [k*64+n]
        int k = (i * 4) / FOUT;
        int n = (i * 4) % FOUT;
        ldsW[(n + 0) * FHID + k] = w.x;
        ldsW[(n + 1) * FHID + k] = w.y;
        ldsW[(n + 2) * FHID + k] = w.z;
        ldsW[(n + 3) * FHID + k] = w.w;
    }
    if (t < FHID) ldsB1[t] = b1[t];
    __syncthreads();

    const int wave = t >> 5;
    const int lane = t & 31;
    const int rowTile = blockIdx.x * 8 + wave;
    const int numRowTiles = M >> 4;
    if (rowTile >= numRowTiles) return;

    const int m = lane & 15;
    const int h = lane >> 4;
    const long row = (long)rowTile * 16 + m;
    const float* hrow = H + row * FHID;

    v8f acc[4];
    v8f zf = {};
#pragma unroll
    for (int nt = 0; nt < 4; ++nt) acc[nt] = zf;

    for (int kk = 0; kk < FHID; kk += 4) {
        const int c0 = kk + 2 * h;
        v2f a;
        a.x = fmaxf(hrow[c0] + ldsB1[c0], 0.0f);          // fused bias + relu
        a.y = fmaxf(hrow[c0 + 1] + ldsB1[c0 + 1], 0.0f);
#pragma unroll
        for (int nt = 0; nt < 4; ++nt) {
            v2f b;
            b.x = ldsW[(nt * 16 + m) * FHID + c0];
            b.y = ldsW[(nt * 16 + m) * FHID + c0 + 1];
            acc[nt] = __builtin_amdgcn_wmma_f32_16x16x4_f32(
                false, a, false, b, (short)0, acc[nt], false, false);
        }
    }

    const long rbase = (long)rowTile * 16 + 8 * h;
#pragma unroll
    for (int nt = 0; nt < 4; ++nt) {
#pragma unroll
        for (int r = 0; r < 8; ++r) {
            out[(rbase + r) * FOUT + nt * 16 + m] = acc[nt][r];
        }
    }
}

// ---------------- edge scatter: out[dst] += feat[src] * dinv[src]*dinv[dst] ----------------
// 1 thread = 4 consecutive features of one (edge or self-loop).
template <int F>
__global__ void __launch_bounds__(256)
scatter_kernel(const float* __restrict__ feat, const int* __restrict__ src,
               const int* __restrict__ dst, const float* __restrict__ dinv,
               float* __restrict__ out, int E, int N) {
    constexpr int LPE = F / 4;  // threads per edge
    long tid = (long)blockIdx.x * blockDim.x + threadIdx.x;
    long total = (long)(E + N) * LPE;
    if (tid >= total) return;

    int e = (int)(tid / LPE);
    int j = (int)(tid % LPE) * 4;
    int s, d;
    if (e < E) { s = src[e]; d = dst[e]; }
    else       { s = d = e - E; }          // self-loop

    float nrm = dinv[s] * dinv[d];
    float4 v = *(const float4*)(feat + (long)s * F + j);
    float* o = out + (long)d * F + j;
    atomicAdd(o + 0, v.x * nrm);
    atomicAdd(o + 1, v.y * nrm);
    atomicAdd(o + 2, v.z * nrm);
    atomicAdd(o + 3, v.w * nrm);
}

// ---------------- host launcher ----------------
extern "C" void kernel_launch(void* const* d_in, const int* in_sizes, int n_in,
                              void* d_out, int out_size, void* d_ws, size_t ws_size,
                              hipStream_t stream) {
    const float* x  = (const float*)d_in[0];
    const int*   ei = (const int*)d_in[1];
    const float* W1 = (const float*)d_in[2];
    const float* b1 = (const float*)d_in[3];
    const float* W2 = (const float*)d_in[4];
    const float* b2 = (const float*)d_in[5];
    float* out = (float*)d_out;

    const int N = in_sizes[0] / FIN;
    const int E = in_sizes[1] / 2;
    const int* src = ei;
    const int* dst = ei + E;

    char* ws = (char*)d_ws;
    float* dinv = (float*)ws;  // N floats
    size_t off = (((size_t)N * 4) + 255) & ~(size_t)255;
    float* XW = (float*)(ws + off);  // N*128 floats; reused as HW2 (N*64) later
    off += (((size_t)N * FHID * 4) + 255) & ~(size_t)255;
    float* H1 = (float*)(ws + off);  // N*128 floats

    const dim3 blk(256);

    // normalization coefficients
    init_deg_kernel<<<(N + 255) / 256, blk, 0, stream>>>(dinv, N);
    deg_accum_kernel<<<(E + 255) / 256, blk, 0, stream>>>(dst, dinv, E);
    dinv_kernel<<<(N + 255) / 256, blk, 0, stream>>>(dinv, N);

    const int rowTiles = N / 16;             // 3125
    const int gBlocks = (rowTiles + 7) / 8;  // 391

    // layer 1
    gemm_x_w1<<<gBlocks, blk, 0, stream>>>(x, W1, XW, N);
    zero4_kernel<<<((N * FHID / 4) + 255) / 256, blk, 0, stream>>>((float4*)H1, N * FHID / 4);
    {
        long total = (long)(E + N) * (FHID / 4);
        scatter_kernel<FHID><<<(int)((total + 255) / 256), blk, 0, stream>>>(
            XW, src, dst, dinv, H1, E, N);
    }

    // layer 2 (bias+relu fused into GEMM2 A-load; out pre-seeded with b2)
    gemm_h_w2<<<gBlocks, blk, 0, stream>>>(H1, b1, W2, XW, N);
    init_out_b2_kernel<<<((N * FOUT) + 255) / 256, blk, 0, stream>>>(out, b2, N * FOUT);
    {
        long total = (long)(E + N) * (FOUT / 4);
        scatter_kernel<FOUT><<<(int)((total + 255) / 256), blk, 0, stream>>>(
            XW, src, dst, dinv, out, E, N);
    }
}